// RoIAlign1D_15874199126293
// MI455X (gfx1250) — compile-verified
//
#include <hip/hip_runtime.h>
#include <stdint.h>

// Problem shape (fixed by the reference): feat [B,T,D], spans [B,K,2], lengths [B]
#define NB 16
#define NT 4096
#define ND 512
#define NK 64
#define NP 16

typedef unsigned int u32x4 __attribute__((ext_vector_type(4)));
typedef unsigned int u32x8 __attribute__((ext_vector_type(8)));
typedef float        f32x4 __attribute__((ext_vector_type(4)));

static __device__ __forceinline__ int imin(int a, int b) { return a < b ? a : b; }
static __device__ __forceinline__ int imax(int a, int b) { return a > b ? a : b; }

// One workgroup per (b,k) RoI. Wave 0 (guarded by a *scalar* branch so waves
// 1..7 cannot re-issue the EXEC-ignoring TDM ops) launches two gather-mode
// tensor_load_to_lds DMAs (16 row indices each == P pooled samples); all 8
// waves then lerp from LDS and stream the 32KB output tile with NT stores.
__global__ __launch_bounds__(256)
void roialign1d_tdm_kernel(const float* __restrict__ feat,
                           const int*   __restrict__ spans,
                           const int*   __restrict__ lengths,
                           float*       __restrict__ out)
{
    __shared__ __attribute__((aligned(16))) float lds0[NP * ND]; // f0 rows, 32KB
    __shared__ __attribute__((aligned(16))) float lds1[NP * ND]; // f1 rows, 32KB

    const int wg = blockIdx.x;        // 0 .. B*K-1  (uniform)
    const int b  = wg / NK;
    const int k  = wg - b * NK;

    // --- uniform RoI metadata (matches reference semantics exactly) ---
    const int Lm1 = lengths[b] - 1;
    int a0 = spans[(b * NK + k) * 2 + 0];
    int a1 = spans[(b * NK + k) * 2 + 1];
    a0 = imin(imax(a0, 0), Lm1);
    a1 = imin(imax(a1, 0), Lm1);
    const int s     = imin(a0, a1);
    const int segm1 = imax(a0, a1) - s;            // seglen - 1  (>= 0)

    // Wave-uniform SGPR condition -> compiler must use s_cmp + s_cbranch
    // (scalar branch), so waves 1..7 branch AROUND the TDM block instead of
    // executing it with EXEC=0. TDM issues regardless of EXEC (ISA 7.1), so
    // an exec-masked guard would duplicate the DMA 8x.
    const int wave_id = __builtin_amdgcn_readfirstlane((int)threadIdx.x) >> 5;

    if (wave_id == 0) {
        // Pack 16-bit row indices (non-decreasing in p, as TDM gather expects).
        unsigned r0[8], r1[8];
#pragma unroll
        for (int i = 0; i < 8; ++i) {
            const int pA = 2 * i, pB = 2 * i + 1;
            const float fA = ((float)pA / 15.0f) * (float)segm1;
            const float fB = ((float)pB / 15.0f) * (float)segm1;
            const int i0A = imin((int)floorf(fA), segm1);
            const int i0B = imin((int)floorf(fB), segm1);
            const int i1A = imin(i0A + 1, segm1);
            const int i1B = imin(i0B + 1, segm1);
            r0[i] = (unsigned)(s + i0A) | ((unsigned)(s + i0B) << 16);
            r1[i] = (unsigned)(s + i1A) | ((unsigned)(s + i1B) << 16);
        }

        const unsigned long long ga =
            (unsigned long long)(uintptr_t)(feat + (size_t)b * (NT * ND));

        // D# group 1 (256b): data_size=4B(code 2)@[17:16]; tensor_dim0=512@[79:48];
        // tensor_dim1=4096@[111:80]; tile_dim0=512@[127:112]; tile_dim1=#idx=16@[143:128];
        // tensor_dim0_stride=512@[207:160]; rest 0.
        u32x8 g1;
        g1[0] = 2u << 16;                  // data_size = 4 bytes
        g1[1] = (unsigned)ND << 16;        // tensor_dim0[15:0]
        g1[2] = (unsigned)NT << 16;        // tensor_dim1[15:0]
        g1[3] = (unsigned)ND << 16;        // tile_dim0
        g1[4] = (unsigned)NP;              // tile_dim1 = #valid gather indices
        g1[5] = (unsigned)ND;              // tensor_dim0_stride (elements)
        g1[6] = 0u;
        g1[7] = 0u;

        // D# group 0 (128b): count=1, gather_mode=1 (bit31), 16-bit indices
        // (bit30=0), lds_addr, 57-bit global_addr, type=2 in bits[127:126].
        u32x4 g0a, g0b, g2a, g3a, g2b, g3b;
        g0a[0] = 0x80000001u;
        g0a[1] = (unsigned)(uintptr_t)(&lds0[0]);  // low 32b of flat = LDS byte addr
        g0a[2] = (unsigned)(ga & 0xFFFFFFFFu);
        g0a[3] = (unsigned)((ga >> 32) & 0x01FFFFFFu) | (2u << 30);
        g0b = g0a;
        g0b[1] = (unsigned)(uintptr_t)(&lds1[0]);

        g2a[0] = r0[0]; g2a[1] = r0[1]; g2a[2] = r0[2]; g2a[3] = r0[3];
        g3a[0] = r0[4]; g3a[1] = r0[5]; g3a[2] = r0[6]; g3a[3] = r0[7];
        g2b[0] = r1[0]; g2b[1] = r1[1]; g2b[2] = r1[2]; g2b[3] = r1[3];
        g3b[0] = r1[4]; g3b[1] = r1[5]; g3b[2] = r1[6]; g3b[3] = r1[7];

        asm volatile("tensor_load_to_lds %0, %1, %2, %3"
                     :: "s"(g0a), "s"(g1), "s"(g2a), "s"(g3a) : "memory");
        asm volatile("tensor_load_to_lds %0, %1, %2, %3"
                     :: "s"(g0b), "s"(g1), "s"(g2b), "s"(g3b) : "memory");
        __builtin_amdgcn_s_wait_tensorcnt(0);
    }
    __syncthreads();

    // --- lerp 16 rows x 512 f32 from LDS, streamed NT b128 stores ---
    const f32x4* l0 = (const f32x4*)lds0;
    const f32x4* l1 = (const f32x4*)lds1;
    f32x4* o = (f32x4*)(out + (size_t)wg * (NP * ND));

#pragma unroll
    for (int j = 0; j < (NP * ND / 4) / 256; ++j) {       // 8 iterations
        const int   vi   = (int)threadIdx.x + j * 256;    // vec4 index, 0..2047
        const int   p    = vi >> 7;                       // 128 vec4 per row
        const float fidx = ((float)p / 15.0f) * (float)segm1;
        const int   i0   = imin((int)floorf(fidx), segm1);
        const float w    = fidx - (float)i0;
        const float wm   = 1.0f - w;

        const f32x4 x = l0[vi];                           // ds_load_b128
        const f32x4 y = l1[vi];
        f32x4 r;
        r.x = wm * x.x + w * y.x;
        r.y = wm * x.y + w * y.y;
        r.z = wm * x.z + w * y.z;
        r.w = wm * x.w + w * y.w;
        __builtin_nontemporal_store(r, &o[vi]);           // global_store_b128 NT
    }
}

extern "C" void kernel_launch(void* const* d_in, const int* in_sizes, int n_in,
                              void* d_out, int out_size, void* d_ws, size_t ws_size,
                              hipStream_t stream) {
    (void)in_sizes; (void)n_in; (void)out_size; (void)d_ws; (void)ws_size;
    const float* feat    = (const float*)d_in[0];
    const int*   spans   = (const int*)d_in[1];
    const int*   lengths = (const int*)d_in[2];
    float*       out     = (float*)d_out;

    dim3 grid(NB * NK);   // 1024 workgroups, one per RoI
    dim3 block(256);      // 8 wave32
    hipLaunchKernelGGL(roialign1d_tdm_kernel, grid, block, 0, stream,
                       feat, spans, lengths, out);
}